// PerfMetric_4844723110657
// MI455X (gfx1250) — compile-verified
//
#include <hip/hip_runtime.h>
#include <math.h>

typedef float v2f __attribute__((ext_vector_type(2)));
typedef float v8f __attribute__((ext_vector_type(8)));

#define TPB 128                       // threads per block = batches per block
#define NPTS 21
#define FPB 63                        // floats per batch (21*3)
#define TILE_FLOATS (TPB * FPB)       // 8064
#define TILE_BYTES  (TILE_FLOATS * 4) // 32256
#define TILE_V4     (TILE_FLOATS / 4) // 2016

// fast HW transcendentals (v_sqrt_f32 / v_rsq_f32 / v_rcp_f32, ~1 ulp)
__device__ __forceinline__ float fsqrt(float x) { return __builtin_amdgcn_sqrtf(x); }
__device__ __forceinline__ float frsq(float x)  { return __builtin_amdgcn_rsqf(x); }
__device__ __forceinline__ float frcp(float x)  { return __builtin_amdgcn_rcpf(x); }

// Full-wave (32 lane) sum using V_WMMA_F32_16X16X4_F32.
// A = per-lane value in VGPR0 (K=0 / K=2 slots), VGPR1 = 0. B = all-ones.
// D is column-constant row-sums; lanes 0-15 hold the sum of one half of the
// rows, lanes 16-31 the other half; xor-16 add completes the reduction.
__device__ __forceinline__ float wave_sum_wmma(float v) {
  v2f a; a[0] = v;    a[1] = 0.0f;
  v2f b; b[0] = 1.0f; b[1] = 1.0f;
  v8f c = {};
  v8f d = __builtin_amdgcn_wmma_f32_16x16x4_f32(false, a, false, b, (short)0, c,
                                                false, false);
  float s = ((d[0] + d[1]) + (d[2] + d[3])) + ((d[4] + d[5]) + (d[6] + d[7]));
  s += __shfl_xor(s, 16);
  return s;
}

// Classical symmetric Jacobi rotation on pair (p,q) with remaining index r.
// app=S[p][p], aqq=S[q][q], apq=S[p][q], arp=S[r][p], arq=S[r][q].
// vp/vq are columns p,q of the accumulated eigenvector matrix V.
__device__ __forceinline__ void jrot(float& app, float& aqq, float& apq,
                                     float& arp, float& arq,
                                     float (&vp)[3], float (&vq)[3]) {
  float tau = (aqq - app) * 0.5f * frcp(apq);
  float t = frcp(fabsf(tau) + fsqrt(fmaf(tau, tau, 1.0f)));
  t = (tau < 0.0f) ? -t : t;
  t = (fabsf(apq) > 1e-30f) ? t : 0.0f;     // kills 0/0 NaN path
  float cc = frsq(fmaf(t, t, 1.0f));
  float ss = t * cc;
  float h = t * apq;
  app -= h;
  aqq += h;
  apq = 0.0f;
  float p0 = arp, q0 = arq;
  arp = cc * p0 - ss * q0;
  arq = ss * p0 + cc * q0;
#pragma unroll
  for (int i = 0; i < 3; ++i) {
    float vpi = vp[i], vqi = vq[i];
    vp[i] = cc * vpi - ss * vqi;
    vq[i] = ss * vpi + cc * vqi;
  }
}

__global__ __launch_bounds__(TPB) void pa_tile_kernel(
    const float* __restrict__ pred,   // Y : [B,21,3]
    const float* __restrict__ targ,   // X : [B,21,3]
    float* __restrict__ partial)
{
  __shared__ __align__(16) float sX[TILE_FLOATS];
  __shared__ __align__(16) float sY[TILE_FLOATS];
  __shared__ float sred[TPB / 32];

  const int t = threadIdx.x;
  const unsigned ldsX  = (unsigned)(size_t)(void*)sX;
  const unsigned ldsY  = (unsigned)(size_t)(void*)sY;
  const unsigned gbase = (unsigned)blockIdx.x * (unsigned)TILE_BYTES;

  // ---- async global -> LDS staging, 16B per lane per issue ----
#pragma unroll
  for (int k = 0; k < (TILE_V4 + TPB - 1) / TPB; ++k) {
    int i4 = t + TPB * k;
    if (i4 < TILE_V4) {
      unsigned go = gbase + (unsigned)i4 * 16u;
      unsigned lx = ldsX + (unsigned)i4 * 16u;
      unsigned ly = ldsY + (unsigned)i4 * 16u;
      asm volatile("global_load_async_to_lds_b128 %0, %1, %2"
                   :: "v"(lx), "v"(go), "s"(targ) : "memory");
      asm volatile("global_load_async_to_lds_b128 %0, %1, %2"
                   :: "v"(ly), "v"(go), "s"(pred) : "memory");
    }
  }
  asm volatile("s_wait_asynccnt 0" ::: "memory");
  __syncthreads();

  // ---- pass 1: means, norms, raw cross-covariance ----
  const int xb = t * FPB;
  float sx0=0.f,sx1=0.f,sx2=0.f, sy0=0.f,sy1=0.f,sy2=0.f, sxx=0.f, syy=0.f;
  float c00=0.f,c01=0.f,c02=0.f,c10=0.f,c11=0.f,c12=0.f,c20=0.f,c21=0.f,c22=0.f;
  for (int n = 0; n < NPTS; ++n) {
    float x0 = sX[xb+3*n+0], x1 = sX[xb+3*n+1], x2 = sX[xb+3*n+2];
    float y0 = sY[xb+3*n+0], y1 = sY[xb+3*n+1], y2 = sY[xb+3*n+2];
    sx0+=x0; sx1+=x1; sx2+=x2; sy0+=y0; sy1+=y1; sy2+=y2;
    sxx = fmaf(x0,x0,fmaf(x1,x1,fmaf(x2,x2,sxx)));
    syy = fmaf(y0,y0,fmaf(y1,y1,fmaf(y2,y2,syy)));
    c00=fmaf(x0,y0,c00); c01=fmaf(x0,y1,c01); c02=fmaf(x0,y2,c02);
    c10=fmaf(x1,y0,c10); c11=fmaf(x1,y1,c11); c12=fmaf(x1,y2,c12);
    c20=fmaf(x2,y0,c20); c21=fmaf(x2,y1,c21); c22=fmaf(x2,y2,c22);
  }
  const float invN = 1.0f / 21.0f;
  float mux0=sx0*invN, mux1=sx1*invN, mux2=sx2*invN;
  float muy0=sy0*invN, muy1=sy1*invN, muy2=sy2*invN;
  float nX2 = sxx - fmaf(sx0,mux0, fmaf(sx1,mux1, sx2*mux2));
  float nY2 = syy - fmaf(sy0,muy0, fmaf(sy1,muy1, sy2*muy2));
  nX2 = fmaxf(nX2, 1e-20f); nY2 = fmaxf(nY2, 1e-20f);
  float rx = frsq(nX2), ry = frsq(nY2);
  float sc = rx * ry;

  // M = X0n^T Y0n (normalized centered cross-covariance)
  float M[3][3];
  M[0][0]=(c00-sx0*muy0)*sc; M[0][1]=(c01-sx0*muy1)*sc; M[0][2]=(c02-sx0*muy2)*sc;
  M[1][0]=(c10-sx1*muy0)*sc; M[1][1]=(c11-sx1*muy1)*sc; M[1][2]=(c12-sx1*muy2)*sc;
  M[2][0]=(c20-sx2*muy0)*sc; M[2][1]=(c21-sx2*muy1)*sc; M[2][2]=(c22-sx2*muy2)*sc;

  // ---- 3x3 SVD via symmetric Jacobi on S = M^T M (6 unique entries) ----
  float s00 = M[0][0]*M[0][0] + M[1][0]*M[1][0] + M[2][0]*M[2][0];
  float s11 = M[0][1]*M[0][1] + M[1][1]*M[1][1] + M[2][1]*M[2][1];
  float s22 = M[0][2]*M[0][2] + M[1][2]*M[1][2] + M[2][2]*M[2][2];
  float s01 = M[0][0]*M[0][1] + M[1][0]*M[1][1] + M[2][0]*M[2][1];
  float s02 = M[0][0]*M[0][2] + M[1][0]*M[1][2] + M[2][0]*M[2][2];
  float s12 = M[0][1]*M[0][2] + M[1][1]*M[1][2] + M[2][1]*M[2][2];
  float vc0[3] = {1.f, 0.f, 0.f};   // columns of V
  float vc1[3] = {0.f, 1.f, 0.f};
  float vc2[3] = {0.f, 0.f, 1.f};
#pragma unroll
  for (int sweep = 0; sweep < 4; ++sweep) {
    jrot(s00, s11, s01, s02, s12, vc0, vc1);   // (p,q)=(0,1), r=2
    jrot(s00, s22, s02, s01, s12, vc0, vc2);   // (p,q)=(0,2), r=1
    jrot(s11, s22, s12, s01, s02, vc1, vc2);   // (p,q)=(1,2), r=0
  }
  float l0 = fmaxf(s00, 0.f), l1 = fmaxf(s11, 0.f), l2 = fmaxf(s22, 0.f);
  float is0 = (l0 > 1e-24f) ? frsq(l0) : 0.0f;
  float is1 = (l1 > 1e-24f) ? frsq(l1) : 0.0f;
  float is2 = (l2 > 1e-24f) ? frsq(l2) : 0.0f;
  float trace = l0*is0 + l1*is1 + l2*is2;      // sum of singular values

  // P = V diag(1/s) V^T ;  Q = V U^T = P M^T
  float P[3][3], Q[3][3];
#pragma unroll
  for (int i = 0; i < 3; ++i)
#pragma unroll
    for (int j = 0; j < 3; ++j)
      P[i][j] = vc0[i]*vc0[j]*is0 + vc1[i]*vc1[j]*is1 + vc2[i]*vc2[j]*is2;
#pragma unroll
  for (int i = 0; i < 3; ++i)
#pragma unroll
    for (int j = 0; j < 3; ++j)
      Q[i][j] = P[i][0]*M[j][0] + P[i][1]*M[j][1] + P[i][2]*M[j][2];

  // ---- pass 2: per-point aligned error ----
  float kfac = trace * nX2 * sc;               // trace * normX / normY
  float esum = 0.0f;
  for (int n = 0; n < NPTS; ++n) {
    float y0 = sY[xb+3*n+0]-muy0, y1 = sY[xb+3*n+1]-muy1, y2 = sY[xb+3*n+2]-muy2;
    float w0 = y0*Q[0][0] + y1*Q[1][0] + y2*Q[2][0];
    float w1 = y0*Q[0][1] + y1*Q[1][1] + y2*Q[2][1];
    float w2 = y0*Q[0][2] + y1*Q[1][2] + y2*Q[2][2];
    float d0 = fmaf(kfac, w0, mux0 - sX[xb+3*n+0]);
    float d1 = fmaf(kfac, w1, mux1 - sX[xb+3*n+1]);
    float d2 = fmaf(kfac, w2, mux2 - sX[xb+3*n+2]);
    esum += fsqrt(fmaf(d0,d0,fmaf(d1,d1,d2*d2)));
  }

  // ---- WMMA wave reduction + cross-wave LDS finish ----
  float wsum = wave_sum_wmma(esum);
  if ((t & 31) == 0) sred[t >> 5] = wsum;
  __syncthreads();
  if (t == 0) {
    float s = 0.0f;
#pragma unroll
    for (int w = 0; w < TPB/32; ++w) s += sred[w];
    partial[blockIdx.x] = s;
  }
}

__global__ __launch_bounds__(256) void final_reduce_kernel(
    const float* __restrict__ partial, int n, float scale,
    float* __restrict__ out)
{
  __shared__ float sm[8];
  float acc = 0.0f;
  for (int i = threadIdx.x; i < n; i += 256) acc += partial[i];
  float wsum = wave_sum_wmma(acc);
  if ((threadIdx.x & 31) == 0) sm[threadIdx.x >> 5] = wsum;
  __syncthreads();
  if (threadIdx.x == 0) {
    float s = 0.0f;
#pragma unroll
    for (int w = 0; w < 8; ++w) s += sm[w];
    out[0] = s * scale;
  }
}

extern "C" void kernel_launch(void* const* d_in, const int* in_sizes, int n_in,
                              void* d_out, int out_size, void* d_ws, size_t ws_size,
                              hipStream_t stream) {
  const float* pred = (const float*)d_in[0];   // pred_kp3d  (Y)
  const float* targ = (const float*)d_in[1];   // target_kp3d (X)
  const int total = in_sizes[0];               // B * 63
  const int B = total / FPB;
  const int nblk = (B + TPB - 1) / TPB;        // 2048 for the reference shape
  float* partial = (float*)d_ws;

  pa_tile_kernel<<<nblk, TPB, 0, stream>>>(pred, targ, partial);
  const float scale = 1.0f / ((float)B * (float)NPTS);
  final_reduce_kernel<<<1, 256, 0, stream>>>(partial, nblk, scale, (float*)d_out);

  (void)n_in; (void)out_size; (void)ws_size;
}